// GaussianCubeAdapter_82841329205489
// MI455X (gfx1250) — compile-verified
//
#include <hip/hip_runtime.h>
#include <math.h>

// ---------------- problem constants (from reference) ----------------
#define B_    4
#define V_    4
#define NPTS  16384
#define CX    64
#define CY    64
#define CZ    128
#define NVOX  (CX*CY*CZ)          // 524288
#define NCH   10                  // 7 + 3*D_SH
#define C0_   0.28209479177387814f
#define EPS_  1e-8f

typedef __attribute__((ext_vector_type(2))) float v2f;
typedef __attribute__((ext_vector_type(8))) float v8f;

#if defined(__AMDGCN__)
typedef __attribute__((address_space(1))) int* as1_int_p;   // global
typedef __attribute__((address_space(3))) int* as3_int_p;   // LDS
#endif

// ---------------- workspace layout (32-bit words) ----------------
// [0, 786432)              : means  (b,v,n,3) f32
// [786432, 786432+262144) : mask bitfield, 4 dwords per (b,v,l,m) row (z fastest)
// [.., +96)                : per-(b,v) min/max, ordered-uint encoded (3 min, 3 max)
#define WS_MEANS_W   0
#define WS_MASK_W    (B_*V_*NPTS*3)
#define N_MASK_WORDS (B_*V_*CX*CY*(CZ/32))
#define WS_MM_W      (WS_MASK_W + N_MASK_WORDS)
#define N_MM         (B_*V_*6)

// monotonic float<->uint encoding so we can use integer atomicMin/Max
__device__ __forceinline__ unsigned enc_f(float f) {
  unsigned u = __float_as_uint(f);
  return (u & 0x80000000u) ? ~u : (u | 0x80000000u);
}
__device__ __forceinline__ float dec_f(unsigned u) {
  return __uint_as_float((u & 0x80000000u) ? (u ^ 0x80000000u) : ~u);
}

// ---------------- kernel 0: clear mask bits, init min/max ----------------
__global__ void k_init(unsigned* __restrict__ mask_w, unsigned* __restrict__ mm) {
  int i = blockIdx.x * 256 + threadIdx.x;
  if (i < N_MASK_WORDS) mask_w[i] = 0u;
  if (i < N_MM)         mm[i] = ((i % 6) < 3) ? 0xFFFFFFFFu : 0u;
}

// ---------------- kernel 1: means + per-(b,v) min/max reduction ----------------
__global__ void k_means(const float* __restrict__ ext, const float* __restrict__ K,
                        const float* __restrict__ coords, const float* __restrict__ depths,
                        float* __restrict__ means_ws, unsigned* __restrict__ mm) {
  int p  = blockIdx.x * 256 + threadIdx.x;     // 0 .. B*V*NPTS-1 (block-uniform bv)
  int bv = p >> 14;
  // general 3x3 inverse of K (matches jnp.linalg.inv)
  const float* Kp = K + bv * 9;
  float a00=Kp[0],a01=Kp[1],a02=Kp[2],a10=Kp[3],a11=Kp[4],a12=Kp[5],a20=Kp[6],a21=Kp[7],a22=Kp[8];
  float det = a00*(a11*a22-a12*a21) - a01*(a10*a22-a12*a20) + a02*(a10*a21-a11*a20);
  float id  = 1.0f / det;
  float i00=(a11*a22-a12*a21)*id, i01=(a02*a21-a01*a22)*id, i02=(a01*a12-a02*a11)*id;
  float i10=(a12*a20-a10*a22)*id, i11=(a00*a22-a02*a20)*id, i12=(a02*a10-a00*a12)*id;
  float i20=(a10*a21-a11*a20)*id, i21=(a01*a20-a00*a21)*id, i22=(a00*a11-a01*a10)*id;

  float u = coords[p*2+0], vv = coords[p*2+1];
  float dx = i00*u + i01*vv + i02;
  float dy = i10*u + i11*vv + i12;
  float dz = i20*u + i21*vv + i22;
  float inv = 1.0f / (sqrtf(dx*dx + dy*dy + dz*dz) + EPS_);
  dx *= inv; dy *= inv; dz *= inv;

  const float* E = ext + bv * 16;              // row-major 4x4
  float wx = E[0]*dx + E[1]*dy + E[2]*dz;
  float wy = E[4]*dx + E[5]*dy + E[6]*dz;
  float wz = E[8]*dx + E[9]*dy + E[10]*dz;
  float d  = depths[p];
  float m0 = E[3] + wx*d, m1 = E[7] + wy*d, m2 = E[11] + wz*d;
  means_ws[p*3+0] = m0; means_ws[p*3+1] = m1; means_ws[p*3+2] = m2;

  // wave32 min/max reduce, then atomics on ordered-uint encodings
  float n0=m0, n1=m1, n2=m2, x0=m0, x1=m1, x2=m2;
  #pragma unroll
  for (int off = 16; off > 0; off >>= 1) {
    n0 = fminf(n0, __shfl_xor(n0, off, 32));
    n1 = fminf(n1, __shfl_xor(n1, off, 32));
    n2 = fminf(n2, __shfl_xor(n2, off, 32));
    x0 = fmaxf(x0, __shfl_xor(x0, off, 32));
    x1 = fmaxf(x1, __shfl_xor(x1, off, 32));
    x2 = fmaxf(x2, __shfl_xor(x2, off, 32));
  }
  if ((threadIdx.x & 31) == 0) {
    atomicMin(&mm[bv*6+0], enc_f(n0));
    atomicMin(&mm[bv*6+1], enc_f(n1));
    atomicMin(&mm[bv*6+2], enc_f(n2));
    atomicMax(&mm[bv*6+3], enc_f(x0));
    atomicMax(&mm[bv*6+4], enc_f(x1));
    atomicMax(&mm[bv*6+5], enc_f(x2));
  }
}

// ---------------- kernel 2: quantize points, scatter mask bits ----------------
__global__ void k_mask(const float* __restrict__ means_ws, const unsigned* __restrict__ mm,
                       unsigned* __restrict__ mask_w) {
  int p  = blockIdx.x * 256 + threadIdx.x;
  int bv = p >> 14;
  int idx[3];
  const float cube[3] = {64.f, 64.f, 128.f};
  #pragma unroll
  for (int c = 0; c < 3; ++c) {
    float mn  = dec_f(mm[bv*6+c]);
    float rng = fmaxf(dec_f(mm[bv*6+3+c]) - mn, 1e-8f);
    float nrm = (means_ws[p*3+c] - mn) / rng * cube[c];
    int ii = (int)rintf(nrm);                 // RNE == jnp.round
    int hi = (c == 2) ? (CZ-1) : (CX-1);
    idx[c] = min(max(ii, 0), hi);
  }
  unsigned widx = (((unsigned)bv*CX + idx[0])*CY + idx[1])*(CZ/32) + (idx[2] >> 5);
  atomicOr(&mask_w[widx], 1u << (idx[2] & 31));
}

// ---------------- kernel 3: means_oracle = sum over views ----------------
__global__ void k_oracle(const float* __restrict__ means_ws, float* __restrict__ o_means) {
  int i = blockIdx.x * 256 + threadIdx.x;      // output element (b*NPTS + pt)*3 + c
  int c = i % 3, t = i / 3;
  int b = t >> 14, pt = t & (NPTS-1);
  float s = 0.f;
  #pragma unroll
  for (int v = 0; v < V_; ++v) s += means_ws[(((b*V_+v) << 14) + pt)*3 + c];
  o_means[i] = s;
}

// ---------------- kernel 4: per-voxel features + WMMA covariance ----------------
__global__ void __launch_bounds__(128)
k_voxel(const float* __restrict__ ext, const float* __restrict__ opacs,
        const float* __restrict__ rawg, const float* __restrict__ imgs,
        const float* __restrict__ shmask, const unsigned* __restrict__ mask_w,
        float* __restrict__ o_cov, float* __restrict__ o_harm, float* __restrict__ o_opac,
        float* __restrict__ o_scales, float* __restrict__ o_rot) {
  __shared__ float s_gx[V_][NCH], s_gy[V_][NCH], s_ox[V_], s_oy[V_];
  const int vx = blockIdx.x >> 6, vy = blockIdx.x & 63;   // (l, m)
  const int b  = blockIdx.y;
  const int nz = threadIdx.x;                              // z / n dimension, 0..127
  const int t  = threadIdx.x;

  // prefetch this thread's gz row (view 0) -> global_prefetch_b8
  __builtin_prefetch(rawg + ((size_t)((b*V_)*256 + 128 + nz))*NCH, 0, 0);

  // stage the 88 broadcast floats (gx/gy rows + ox/oy) into LDS.
  if (t < 88) {
    const float* src; float* dst;
    if (t < 40)      { int v = t/10,  k = t-10*v;  src = rawg + ((b*V_+v)*256 + vx)*NCH + k;      dst = &s_gx[v][k]; }
    else if (t < 80) { int tt = t-40, v = tt/10, k = tt-10*v; src = rawg + ((b*V_+v)*256 + 64+vy)*NCH + k; dst = &s_gy[v][k]; }
    else if (t < 84) { int v = t-80;  src = opacs + (b*V_+v)*256 + vx;      dst = &s_ox[v]; }
    else             { int v = t-84;  src = opacs + (b*V_+v)*256 + 64 + vy; dst = &s_oy[v]; }
#if __has_builtin(__builtin_amdgcn_global_load_async_to_lds_b32)
    __builtin_amdgcn_global_load_async_to_lds_b32(
        (as1_int_p)(void*)src, (as3_int_p)(void*)dst, 0, 0);
#else
    *dst = *src;
#endif
  }
#if __has_builtin(__builtin_amdgcn_global_load_async_to_lds_b32)
  asm volatile("s_wait_asynccnt 0" ::: "memory");
#endif
  __syncthreads();

  // ---- masked triple-product accumulation over views ----
  float g[NCH];
  #pragma unroll
  for (int k = 0; k < NCH; ++k) g[k] = 0.f;
  float osum = 0.f;
  const unsigned wsel = (unsigned)nz >> 5;
  const unsigned bit  = 1u << (nz & 31);
  #pragma unroll
  for (int v = 0; v < V_; ++v) {
    unsigned w = mask_w[(((b*V_+v)*CX + vx)*CY + vy)*(CZ/32) + wsel];
    if (w & bit) {
      const float* gz = rawg + ((b*V_+v)*256 + 128 + nz)*NCH;
      #pragma unroll
      for (int k = 0; k < NCH; ++k) g[k] += s_gx[v][k]*s_gy[v][k]*gz[k];
      osum += s_ox[v]*s_oy[v]*opacs[(b*V_+v)*256 + 128 + nz];
    }
  }

  const int vox = (vx*CY + vy)*CZ + nz;
  const size_t base = (size_t)b*NVOX + vox;

  // opacity: sigmoid
  o_opac[base] = 1.f / (1.f + expf(-osum));

  // scales: clip(softplus(g - 4), 0.5, 15)
  float sc[3];
  #pragma unroll
  for (int c = 0; c < 3; ++c) {
    float xx = g[c] - 4.f;
    float sp = (xx > 20.f) ? xx : log1pf(expf(xx));
    sc[c] = fminf(fmaxf(sp, 0.5f), 15.f);
    o_scales[base*3 + c] = sc[c];
  }

  // rot: normalized quaternion
  float qw = g[3], qx = g[4], qy = g[5], qz = g[6];
  float qi = 1.f / (sqrtf(qw*qw + qx*qx + qy*qy + qz*qz) + EPS_);
  qw *= qi; qx *= qi; qy *= qi; qz *= qi;
  o_rot[base*4+0] = qw; o_rot[base*4+1] = qx; o_rot[base*4+2] = qy; o_rot[base*4+3] = qz;

  // harmonics: sh * mask + (img - 0.5)/C0 on DC term
  float shm = shmask[0];
  #pragma unroll
  for (int c = 0; c < 3; ++c)
    o_harm[base*3 + c] = g[7+c]*shm + (imgs[b*V_*3 + c] - 0.5f) * (1.f / C0_);

  // ---- local covariance Sigma = (R*diag(s)) (R*diag(s))^T (symmetric) ----
  float r00=1.f-2.f*(qy*qy+qz*qz), r01=2.f*(qx*qy-qw*qz), r02=2.f*(qx*qz+qw*qy);
  float r10=2.f*(qx*qy+qw*qz), r11=1.f-2.f*(qx*qx+qz*qz), r12=2.f*(qy*qz-qw*qx);
  float r20=2.f*(qx*qz-qw*qy), r21=2.f*(qy*qz+qw*qx), r22=1.f-2.f*(qx*qx+qy*qy);
  float A00=r00*sc[0],A01=r01*sc[1],A02=r02*sc[2];
  float A10=r10*sc[0],A11=r11*sc[1],A12=r12*sc[2];
  float A20=r20*sc[0],A21=r21*sc[1],A22=r22*sc[2];
  float s00=A00*A00+A01*A01+A02*A02;
  float s01=A00*A10+A01*A11+A02*A12;
  float s02=A00*A20+A01*A21+A02*A22;
  float s11=A10*A10+A11*A11+A12*A12;
  float s12=A10*A20+A11*A21+A12*A22;
  float s22=A20*A20+A21*A21+A22*A22;

  const float* E = ext + b*V_*16;  // view 0 extrinsics; C = c2w = E[:3,:3]

#if __has_builtin(__builtin_amdgcn_wmma_f32_16x16x4_f32)
  // Batched congruence C*Sigma*C^T via exact-f32 WMMA (16x16x4).
  // A (16x4) = C zero-padded.  Documented A layout: lane m<16 holds row m,
  // VGPR0=K0/VGPR1=K1; lanes 16-31 hold K2/K3.
  // B (4x16) layout (mirrors C/D half-split): VGPR0 = rows K0 (lanes 0-15) /
  // K2 (lanes 16-31); VGPR1 = rows K1 / K3.
  // D layout (documented): lane j<16, VGPR r = D[r][j].
  const unsigned lane = threadIdx.x & 31u;
  const unsigned l15  = lane & 15u;
  const bool hi = lane >= 16u;
  float a0 = 0.f, a1 = 0.f;
  if (l15 < 3u) { a0 = hi ? E[l15*4+2] : E[l15*4+0]; a1 = hi ? 0.f : E[l15*4+1]; }
  v2f Amat = {a0, a1};
  const int wavebase = (int)(threadIdx.x & ~31u);
  const v8f zc = {0.f,0.f,0.f,0.f,0.f,0.f,0.f,0.f};
  #pragma unroll
  for (int grp = 0; grp < 2; ++grp) {          // 16 voxels per WMMA group
    int srcl = grp*16 + (int)l15;              // owner lane of column (lane&15)
    float t00=__shfl(s00,srcl,32), t01=__shfl(s01,srcl,32), t02=__shfl(s02,srcl,32);
    float t11=__shfl(s11,srcl,32), t12=__shfl(s12,srcl,32), t22=__shfl(s22,srcl,32);
    // pass 1: M = C * Sigma, one WMMA per Sigma column c
    v2f B0 = { hi ? t02 : t00, hi ? 0.f : t01 };
    v8f D0 = __builtin_amdgcn_wmma_f32_16x16x4_f32(false, Amat, false, B0, (short)0, zc, false, false);
    v2f B1 = { hi ? t12 : t01, hi ? 0.f : t11 };
    v8f D1 = __builtin_amdgcn_wmma_f32_16x16x4_f32(false, Amat, false, B1, (short)0, zc, false, false);
    v2f B2 = { hi ? t22 : t02, hi ? 0.f : t12 };
    v8f D2 = __builtin_amdgcn_wmma_f32_16x16x4_f32(false, Amat, false, B2, (short)0, zc, false, false);
    // pass 2: out[r][:] = C * M[r][:]^T  (M[r][c] = D_c[r] in lane j)
    int nn = wavebase + grp*16 + (int)l15;
    int vvox = (vx*CY + vy)*CZ + nn;
    float* cp = o_cov + ((size_t)b*NVOX + vvox)*9;
    #pragma unroll
    for (int r = 0; r < 3; ++r) {
      float mr0 = D0[r], mr1 = D1[r], mr2 = D2[r];
      float t2  = __shfl(mr2, (int)l15, 32);   // row K2 values come from lane j
      v2f Br = { hi ? t2 : mr0, hi ? 0.f : mr1 };
      v8f Dp = __builtin_amdgcn_wmma_f32_16x16x4_f32(false, Amat, false, Br, (short)0, zc, false, false);
      if (!hi) { cp[r*3+0] = Dp[0]; cp[r*3+1] = Dp[1]; cp[r*3+2] = Dp[2]; }
    }
  }
#else
  // scalar fallback: out = C * Sigma * C^T
  float Cm[3][3] = {{E[0],E[1],E[2]},{E[4],E[5],E[6]},{E[8],E[9],E[10]}};
  float S[3][3]  = {{s00,s01,s02},{s01,s11,s12},{s02,s12,s22}};
  float M[3][3];
  #pragma unroll
  for (int i = 0; i < 3; ++i)
    #pragma unroll
    for (int j = 0; j < 3; ++j)
      M[i][j] = Cm[i][0]*S[0][j] + Cm[i][1]*S[1][j] + Cm[i][2]*S[2][j];
  float* cp = o_cov + base*9;
  #pragma unroll
  for (int i = 0; i < 3; ++i)
    #pragma unroll
    for (int j = 0; j < 3; ++j)
      cp[i*3+j] = M[i][0]*Cm[j][0] + M[i][1]*Cm[j][1] + M[i][2]*Cm[j][2];
#endif
}

// ---------------- host launcher ----------------
extern "C" void kernel_launch(void* const* d_in, const int* in_sizes, int n_in,
                              void* d_out, int out_size, void* d_ws, size_t ws_size,
                              hipStream_t stream) {
  (void)in_sizes; (void)n_in; (void)out_size; (void)ws_size;
  const float* ext    = (const float*)d_in[0];
  const float* K      = (const float*)d_in[1];
  const float* coords = (const float*)d_in[2];
  const float* depths = (const float*)d_in[3];
  const float* opacs  = (const float*)d_in[4];
  const float* rawg   = (const float*)d_in[5];
  const float* imgs   = (const float*)d_in[6];
  // d_in[7] = coef: consumed only by dead code in the reference
  const float* shm    = (const float*)d_in[8];

  float* out       = (float*)d_out;
  float* o_means   = out;                                   // (4,1,16384,1,1,3)
  float* o_cov     = o_means  + (size_t)B_*NPTS*3;          // (4,NVOX,1,1,3,3)
  float* o_harm    = o_cov    + (size_t)B_*NVOX*9;          // (4,NVOX,1,1,3,1)
  float* o_opac    = o_harm   + (size_t)B_*NVOX*3;          // (4,NVOX,1,1)
  float* o_scales  = o_opac   + (size_t)B_*NVOX;            // (4,NVOX,1,1,3)
  float* o_rot     = o_scales + (size_t)B_*NVOX*3;          // (4,NVOX,1,1,4)

  float*    means_ws = (float*)d_ws;
  unsigned* mask_w   = (unsigned*)d_ws + WS_MASK_W;
  unsigned* mm       = (unsigned*)d_ws + WS_MM_W;

  k_init  <<<dim3((N_MASK_WORDS + 255)/256), 256, 0, stream>>>(mask_w, mm);
  k_means <<<dim3(B_*V_*NPTS/256), 256, 0, stream>>>(ext, K, coords, depths, means_ws, mm);
  k_mask  <<<dim3(B_*V_*NPTS/256), 256, 0, stream>>>(means_ws, mm, mask_w);
  k_oracle<<<dim3(B_*NPTS*3/256), 256, 0, stream>>>(means_ws, o_means);
  k_voxel <<<dim3(CX*CY, B_), 128, 0, stream>>>(ext, opacs, rawg, imgs, shm, mask_w,
                                                o_cov, o_harm, o_opac, o_scales, o_rot);
}